// CalibrationNetwork_44985487458585
// MI455X (gfx1250) — compile-verified
//
#include <hip/hip_runtime.h>

// Problem constants
#define B_   32768
#define D_   35
#define DP   36      // D+1 (K of layer 1, multiple of 4)
#define H_   256
#define K2_  257     // H+1
#define K2P  260     // padded to multiple of 4
#define J_   12
#define Q_   7
#define O_   5
#define NOUT 35      // Q*O
#define NP   48      // padded head width (3 tiles of 16)
#define TM   64      // rows per workgroup tile (4 M-blocks of 16)
#define MAXT (B_/TM + J_)

typedef __attribute__((ext_vector_type(2))) float v2f;
typedef __attribute__((ext_vector_type(8))) float v8f;

// pair-interleaved transposed weight layout: element (k, n) of a (K x N) B-matrix
// stored at (k>>1)*(2*N) + 2*n + (k&1)  -> per-lane b64 load yields {B[k][n], B[k+1][n]}

// ---- weight combine kernels ----
__global__ void calib_w1(const float* __restrict__ W1w, const float* __restrict__ W1aw,
                         float* __restrict__ w1) {
  int i = blockIdx.x * 256 + threadIdx.x;
  if (i >= J_ * DP * H_) return;
  int j = i / (DP * H_), r = i % (DP * H_);
  int k = r / H_, h = r % H_;
  float v = W1w[h * DP + k] + W1aw[(j * H_ + h) * DP + k];
  w1[j * DP * H_ + (k >> 1) * (2 * H_) + h * 2 + (k & 1)] = v;
}

__global__ void calib_w2(const float* __restrict__ W2w, const float* __restrict__ W2aw,
                         float* __restrict__ w2) {
  int i = blockIdx.x * 256 + threadIdx.x;
  if (i >= J_ * K2P * H_) return;
  int j = i / (K2P * H_), r = i % (K2P * H_);
  int k = r / H_, h = r % H_;
  float v = 0.0f;
  if (k < K2_) v = W2w[h * K2_ + k] + W2aw[(j * H_ + h) * K2_ + k];
  w2[j * K2P * H_ + (k >> 1) * (2 * H_) + h * 2 + (k & 1)] = v;
}

__global__ void calib_wv(const float* __restrict__ Vw, const float* __restrict__ Vaw,
                         float* __restrict__ wv) {
  int i = blockIdx.x * 256 + threadIdx.x;
  if (i >= J_ * K2P * NP) return;
  int j = i / (K2P * NP), r = i % (K2P * NP);
  int k = r / NP, n = r % NP;
  float v = 0.0f;
  if (k < K2_ && n < NOUT) v = Vw[n * K2_ + k] + Vaw[(j * NOUT + n) * K2_ + k];
  wv[j * K2P * NP + (k >> 1) * (2 * NP) + n * 2 + (k & 1)] = v;
}

__global__ void calib_bias(const float* __restrict__ W1b, const float* __restrict__ W1ab,
                           const float* __restrict__ W2b, const float* __restrict__ W2ab,
                           const float* __restrict__ Vb,  const float* __restrict__ Vab,
                           float* __restrict__ b1, float* __restrict__ b2,
                           float* __restrict__ bv) {
  int i = blockIdx.x * 256 + threadIdx.x;
  int n1 = J_ * H_;
  if (i < n1) {
    b1[i] = W1b[i % H_] + W1ab[i];
  } else if (i < 2 * n1) {
    int k = i - n1;
    b2[k] = W2b[k % H_] + W2ab[k];
  } else if (i < 2 * n1 + J_ * NP) {
    int k = i - 2 * n1;
    int j = k / NP, n = k % NP;
    bv[k] = (n < NOUT) ? (Vb[n] + Vab[j * NOUT + n]) : 0.0f;
  }
}

// ---- binning kernels ----
__global__ void calib_zero(int* __restrict__ counts, int* __restrict__ counters) {
  int i = threadIdx.x;
  if (i < J_) { counts[i] = 0; counters[i] = 0; }
}

__global__ void calib_count(const int* __restrict__ ids, int* __restrict__ counts) {
  int b = blockIdx.x * 256 + threadIdx.x;
  if (b < B_) atomicAdd(&counts[ids[b]], 1);
}

__global__ void calib_build(const int* __restrict__ counts, int* __restrict__ offs,
                            int* __restrict__ descJ, int* __restrict__ descStart,
                            int* __restrict__ descRows, int* __restrict__ ntiles) {
  if (threadIdx.x != 0 || blockIdx.x != 0) return;
  int total = 0, t = 0;
  for (int j = 0; j < J_; ++j) {
    offs[j] = total;
    int c = counts[j];
    for (int s = 0; s < c; s += TM) {
      descJ[t] = j;
      descStart[t] = total + s;
      descRows[t] = (c - s < TM) ? (c - s) : TM;
      ++t;
    }
    total += c;
  }
  ntiles[0] = t;
}

__global__ void calib_scatter(const int* __restrict__ ids, const int* __restrict__ offs,
                              int* __restrict__ counters, int* __restrict__ perm) {
  int b = blockIdx.x * 256 + threadIdx.x;
  if (b >= B_) return;
  int j = ids[b];
  int pos = offs[j] + atomicAdd(&counters[j], 1);
  perm[pos] = b;
}

// ---- fused MLP kernel: one 64-row tile per workgroup, 8 waves ----
__global__ __launch_bounds__(256) void calib_main(
    const float* __restrict__ x, const int* __restrict__ perm,
    const int* __restrict__ descJ, const int* __restrict__ descStart,
    const int* __restrict__ descRows, const int* __restrict__ ntiles,
    const float* __restrict__ w1, const float* __restrict__ w2,
    const float* __restrict__ wv, const float* __restrict__ b1,
    const float* __restrict__ b2, const float* __restrict__ bv,
    float* __restrict__ out) {
  // LDS: [ xb (TM*DP=2304) ] aliased with [ lg (TM*NP=3072) ],
  // then z1 (TM*K2P=16640), z2 (16640) -> 36352 floats = 145.4 KB
  __shared__ float smem[TM * NP + 2 * TM * K2P];
  float* xb = smem;                 // TM*DP, dead after layer 1
  float* lg = smem;                 // TM*NP, written in layer 3
  float* z1 = smem + TM * NP;       // TM*K2P
  float* z2 = z1 + TM * K2P;        // TM*K2P

  int t = blockIdx.x;
  if (t >= ntiles[0]) return;
  int rows = descRows[t];
  int j = descJ[t];
  int start = descStart[t];

  int tid = threadIdx.x;
  int lane = tid & 31;
  int wave = tid >> 5;
  int hl = lane & 15;           // N (B/C/D frags) or M (A frag) within 16
  int kh = (lane >> 4) << 1;    // +0 for lanes 0-15, +2 for lanes 16-31
  int rb = (lane >> 4) * 8;     // C/D row base

  // stage xb tile into LDS (append bias column = 1, pad rows = 0)
  for (int i = tid; i < TM * DP; i += 256) {
    int r = i / DP, d = i % DP;
    float v = 0.0f;
    if (r < rows) v = (d < D_) ? x[(size_t)perm[start + r] * D_ + d] : 1.0f;
    xb[i] = v;
  }
  __syncthreads();

  // ---- layer 1: (64 x 36) @ (36 x 256) ----
  {
    int n0 = wave * 32;
    v8f acc[4][2];
#pragma unroll
    for (int m = 0; m < 4; ++m) { acc[m][0] = (v8f){}; acc[m][1] = (v8f){}; }
    const float* bw = w1 + (size_t)j * DP * H_;
    for (int k0 = 0; k0 < DP; k0 += 4) {
      int kb = k0 + kh;
      const float* p = bw + (size_t)(kb >> 1) * (2 * H_) + (n0 + hl) * 2;
      v2f bf0 = *(const v2f*)p;
      v2f bf1 = *(const v2f*)(p + 32);
#pragma unroll
      for (int m = 0; m < 4; ++m) {
        v2f a = *(const v2f*)&xb[(m * 16 + hl) * DP + kb];
        acc[m][0] = __builtin_amdgcn_wmma_f32_16x16x4_f32(false, a, false, bf0, (short)0, acc[m][0], false, false);
        acc[m][1] = __builtin_amdgcn_wmma_f32_16x16x4_f32(false, a, false, bf1, (short)0, acc[m][1], false, false);
      }
    }
    const float* bias = b1 + j * H_;
    float bc0 = bias[n0 + hl], bc1 = bias[n0 + 16 + hl];
#pragma unroll
    for (int m = 0; m < 4; ++m) {
#pragma unroll
      for (int v = 0; v < 8; ++v) {
        int r = m * 16 + rb + v;
        float a = acc[m][0][v] + bc0, b = acc[m][1][v] + bc1;
        z1[r * K2P + n0 + hl]      = a > 0.0f ? a : 0.0f;
        z1[r * K2P + n0 + 16 + hl] = b > 0.0f ? b : 0.0f;
      }
    }
  }
  { int r = tid >> 2, c = H_ + (tid & 3); z1[r * K2P + c] = (c == H_) ? 1.0f : 0.0f; }
  __syncthreads();

  // ---- layer 2: (64 x 260) @ (260 x 256) ----
  {
    int n0 = wave * 32;
    v8f acc[4][2];
#pragma unroll
    for (int m = 0; m < 4; ++m) { acc[m][0] = (v8f){}; acc[m][1] = (v8f){}; }
    const float* bw = w2 + (size_t)j * K2P * H_;
    for (int k0 = 0; k0 < K2P; k0 += 4) {
      int kb = k0 + kh;
      const float* p = bw + (size_t)(kb >> 1) * (2 * H_) + (n0 + hl) * 2;
      v2f bf0 = *(const v2f*)p;
      v2f bf1 = *(const v2f*)(p + 32);
#pragma unroll
      for (int m = 0; m < 4; ++m) {
        v2f a = *(const v2f*)&z1[(m * 16 + hl) * K2P + kb];
        acc[m][0] = __builtin_amdgcn_wmma_f32_16x16x4_f32(false, a, false, bf0, (short)0, acc[m][0], false, false);
        acc[m][1] = __builtin_amdgcn_wmma_f32_16x16x4_f32(false, a, false, bf1, (short)0, acc[m][1], false, false);
      }
    }
    const float* bias = b2 + j * H_;
    float bc0 = bias[n0 + hl], bc1 = bias[n0 + 16 + hl];
#pragma unroll
    for (int m = 0; m < 4; ++m) {
#pragma unroll
      for (int v = 0; v < 8; ++v) {
        int r = m * 16 + rb + v;
        float a = acc[m][0][v] + bc0, b = acc[m][1][v] + bc1;
        z2[r * K2P + n0 + hl]      = a > 0.0f ? a : 0.0f;
        z2[r * K2P + n0 + 16 + hl] = b > 0.0f ? b : 0.0f;
      }
    }
  }
  { int r = tid >> 2, c = H_ + (tid & 3); z2[r * K2P + c] = (c == H_) ? 1.0f : 0.0f; }
  __syncthreads();

  // ---- heads: (64 x 260) @ (260 x 48) on waves 0..5 ----
  if (wave < 6) {
    int n0 = (wave % 3) * 16;
    int m0 = (wave / 3) * 32;   // each wave covers 2 M-blocks, reusing B
    v8f acc0 = {}, acc1 = {};
    const float* bw = wv + (size_t)j * K2P * NP;
    for (int k0 = 0; k0 < K2P; k0 += 4) {
      int kb = k0 + kh;
      const float* p = bw + (size_t)(kb >> 1) * (2 * NP) + (n0 + hl) * 2;
      v2f bf = *(const v2f*)p;
      v2f a0 = *(const v2f*)&z2[(m0 + hl) * K2P + kb];
      v2f a1 = *(const v2f*)&z2[(m0 + 16 + hl) * K2P + kb];
      acc0 = __builtin_amdgcn_wmma_f32_16x16x4_f32(false, a0, false, bf, (short)0, acc0, false, false);
      acc1 = __builtin_amdgcn_wmma_f32_16x16x4_f32(false, a1, false, bf, (short)0, acc1, false, false);
    }
    float bc = bv[j * NP + n0 + hl];
#pragma unroll
    for (int v = 0; v < 8; ++v) {
      lg[(m0 + rb + v) * NP + n0 + hl]      = acc0[v] + bc;
      lg[(m0 + 16 + rb + v) * NP + n0 + hl] = acc1[v] + bc;
    }
  }
  __syncthreads();

  // ---- softmax over O=5 per (row, q), scatter to original row ----
  for (int i = tid; i < TM * Q_; i += 256) {
    int r = i / Q_, q = i % Q_;
    if (r < rows) {
      const float* lp = &lg[r * NP + q * O_];
      float m = lp[0];
#pragma unroll
      for (int o = 1; o < O_; ++o) m = fmaxf(m, lp[o]);
      float e[O_], s = 0.0f;
#pragma unroll
      for (int o = 0; o < O_; ++o) { e[o] = __expf(lp[o] - m); s += e[o]; }
      float inv = 1.0f / s;
      float* op = out + (size_t)perm[start + r] * NOUT + q * O_;
#pragma unroll
      for (int o = 0; o < O_; ++o) op[o] = e[o] * inv;
    }
  }
}

extern "C" void kernel_launch(void* const* d_in, const int* in_sizes, int n_in,
                              void* d_out, int out_size, void* d_ws, size_t ws_size,
                              hipStream_t stream) {
  (void)in_sizes; (void)n_in; (void)out_size; (void)ws_size;
  const float* x    = (const float*)d_in[0];
  const int*   ids  = (const int*)d_in[1];
  const float* W1w  = (const float*)d_in[2];
  const float* W1b  = (const float*)d_in[3];
  const float* W2w  = (const float*)d_in[4];
  const float* W2b  = (const float*)d_in[5];
  const float* W1aw = (const float*)d_in[6];
  const float* W1ab = (const float*)d_in[7];
  const float* W2aw = (const float*)d_in[8];
  const float* W2ab = (const float*)d_in[9];
  const float* Vw   = (const float*)d_in[10];
  const float* Vb   = (const float*)d_in[11];
  const float* Vaw  = (const float*)d_in[12];
  const float* Vab  = (const float*)d_in[13];

  float* ws = (float*)d_ws;
  float* w1 = ws;                              // J*DP*H
  float* w2 = w1 + J_ * DP * H_;               // J*K2P*H
  float* wv = w2 + J_ * K2P * H_;              // J*K2P*NP
  float* b1 = wv + J_ * K2P * NP;              // J*H
  float* b2 = b1 + J_ * H_;                    // J*H
  float* bvv = b2 + J_ * H_;                   // J*NP
  int* ib        = (int*)(bvv + J_ * NP);
  int* perm      = ib;                         // B_
  int* descJ     = perm + B_;                  // MAXT
  int* descStart = descJ + MAXT;               // MAXT
  int* descRows  = descStart + MAXT;           // MAXT
  int* offs      = descRows + MAXT;            // J
  int* counts    = offs + J_;                  // J
  int* counters  = counts + J_;                // J
  int* ntl       = counters + J_;              // 1

  calib_w1<<<(J_ * DP * H_ + 255) / 256, 256, 0, stream>>>(W1w, W1aw, w1);
  calib_w2<<<(J_ * K2P * H_ + 255) / 256, 256, 0, stream>>>(W2w, W2aw, w2);
  calib_wv<<<(J_ * K2P * NP + 255) / 256, 256, 0, stream>>>(Vw, Vaw, wv);
  calib_bias<<<(J_ * (2 * H_ + NP) + 255) / 256, 256, 0, stream>>>(W1b, W1ab, W2b, W2ab, Vb, Vab, b1, b2, bvv);
  calib_zero<<<1, 64, 0, stream>>>(counts, counters);
  calib_count<<<B_ / 256, 256, 0, stream>>>(ids, counts);
  calib_build<<<1, 1, 0, stream>>>(counts, offs, descJ, descStart, descRows, ntl);
  calib_scatter<<<B_ / 256, 256, 0, stream>>>(ids, offs, counters, perm);
  calib_main<<<MAXT, 256, 0, stream>>>(x, perm, descJ, descStart, descRows, ntl,
                                       w1, w2, wv, b1, b2, bvv, (float*)d_out);
}